// EarthMoverDistance_48498770707007
// MI455X (gfx1250) — compile-verified
//
#include <hip/hip_runtime.h>

// Earth Mover Distance (approx auction match + cost), MI455X / gfx1250.
// f32 end-to-end (exp(level*d2) with level down to -16384 forbids low precision).
// Pairwise d2 cross-term computed with V_WMMA_F32_16X16X4_F32 per 16x16 tile.

typedef __attribute__((ext_vector_type(2))) float v2f;
typedef __attribute__((ext_vector_type(8))) float v8f;

#define BB 8
#define NN 2048
#define TILES (NN / 16)          // 128 tiles per dimension
#define STRIPS (BB * (NN / 16))  // 1024 wave-strips total

__device__ __forceinline__ v8f dot_tile(v2f a, v2f b) {
  v8f c = {};
  // D = A(16x4 f32) x B(4x16 f32) + 0  -> v_wmma_f32_16x16x4_f32
  return __builtin_amdgcn_wmma_f32_16x16x4_f32(
      /*neg_a=*/false, a, /*neg_b=*/false, b,
      /*c_mod=*/(short)0, c, /*reuse_a=*/false, /*reuse_b=*/false);
}

// ---- Pass 1: rowsum[b,n] = sum_m exp(level*d2)*remR[m] --------------------
__global__ void __launch_bounds__(256)
k_rowsum(const float* __restrict__ xyz1, const float* __restrict__ xyz2,
         const float* __restrict__ remR, float* __restrict__ rowsum,
         float level) {
  const int wid  = (blockIdx.x * 256 + threadIdx.x) >> 5;  // 0..1023
  const int lane = threadIdx.x & 31;
  const int lo = lane & 15, hi = lane >> 4;
  const int b = wid >> 7, strip = wid & 127;

  const int row = strip * 16 + lo;
  const float* p1 = xyz1 + (b * NN + row) * 3;
  const float x1 = p1[0], y1 = p1[1], z1 = p1[2];
  v2f a; a.x = hi ? z1 : x1; a.y = hi ? 0.0f : y1;
  const float n2 = x1 * x1 + y1 * y1 + z1 * z1;
  float n2row[8];
#pragma unroll
  for (int g = 0; g < 8; ++g) n2row[g] = __shfl(n2, g + 8 * hi);

  float acc[8] = {0, 0, 0, 0, 0, 0, 0, 0};
  const float* x2b = xyz2 + b * NN * 3;
  const float* rb  = remR + b * NN;
  for (int t = 0; t < TILES; ++t) {
    const int col = t * 16 + lo;
    const float* p2 = x2b + col * 3;
    const float x2 = p2[0], y2 = p2[1], z2 = p2[2];
    const float m2 = x2 * x2 + y2 * y2 + z2 * z2;
    const float rr = rb[col];
    v2f bv; bv.x = hi ? z2 : x2; bv.y = hi ? 0.0f : y2;
    v8f d = dot_tile(a, bv);
#pragma unroll
    for (int g = 0; g < 8; ++g) {
      float d2 = fmaxf(n2row[g] + m2 - 2.0f * d[g], 0.0f);
      acc[g] += __expf(level * d2) * rr;
    }
  }
#pragma unroll
  for (int g = 0; g < 8; ++g) {
    acc[g] += __shfl_xor(acc[g], 1);
    acc[g] += __shfl_xor(acc[g], 2);
    acc[g] += __shfl_xor(acc[g], 4);
    acc[g] += __shfl_xor(acc[g], 8);
  }
  if (lo == 0) {
#pragma unroll
    for (int g = 0; g < 8; ++g)
      rowsum[b * NN + strip * 16 + g + 8 * hi] = acc[g];
  }
}

// ---- Pass 2: colsum[b,m] = sum_n exp(level*d2)*remR[m]*Lfac[n] ------------
__global__ void __launch_bounds__(256)
k_colsum(const float* __restrict__ xyz1, const float* __restrict__ xyz2,
         const float* __restrict__ remR, const float* __restrict__ Lfac,
         float* __restrict__ colsum, float level) {
  const int wid  = (blockIdx.x * 256 + threadIdx.x) >> 5;
  const int lane = threadIdx.x & 31;
  const int lo = lane & 15, hi = lane >> 4;
  const int b = wid >> 7, strip = wid & 127;

  const int col = strip * 16 + lo;
  const float* p2 = xyz2 + (b * NN + col) * 3;
  const float x2 = p2[0], y2 = p2[1], z2 = p2[2];
  v2f bv; bv.x = hi ? z2 : x2; bv.y = hi ? 0.0f : y2;
  const float m2 = x2 * x2 + y2 * y2 + z2 * z2;
  const float rr = remR[b * NN + col];

  float cacc = 0.0f;
  const float* x1b = xyz1 + b * NN * 3;
  const float* lfb = Lfac + b * NN;
  for (int t = 0; t < TILES; ++t) {
    const int row = t * 16 + lo;
    const float* p1 = x1b + row * 3;
    const float x1 = p1[0], y1 = p1[1], z1 = p1[2];
    v2f a; a.x = hi ? z1 : x1; a.y = hi ? 0.0f : y1;
    const float n2 = x1 * x1 + y1 * y1 + z1 * z1;
    const float lf = lfb[row];
    float n2row[8], lfrow[8];
#pragma unroll
    for (int g = 0; g < 8; ++g) {
      n2row[g] = __shfl(n2, g + 8 * hi);
      lfrow[g] = __shfl(lf, g + 8 * hi);
    }
    v8f d = dot_tile(a, bv);
#pragma unroll
    for (int g = 0; g < 8; ++g) {
      float d2 = fmaxf(n2row[g] + m2 - 2.0f * d[g], 0.0f);
      cacc += __expf(level * d2) * rr * lfrow[g];
    }
  }
  cacc += __shfl_down(cacc, 16);
  if (lane < 16) colsum[b * NN + strip * 16 + lane] = cacc;
}

// ---- Pass 3a: per-row w sums + cost; remainL update (in place) ------------
__global__ void __launch_bounds__(256)
k_update_rows(const float* __restrict__ xyz1, const float* __restrict__ xyz2,
              const float* __restrict__ remR, const float* __restrict__ Lfac,
              const float* __restrict__ fac, float* __restrict__ remL,
              float* __restrict__ costRow, float level) {
  const int wid  = (blockIdx.x * 256 + threadIdx.x) >> 5;
  const int lane = threadIdx.x & 31;
  const int lo = lane & 15, hi = lane >> 4;
  const int b = wid >> 7, strip = wid & 127;

  const int row = strip * 16 + lo;
  const float* p1 = xyz1 + (b * NN + row) * 3;
  const float x1 = p1[0], y1 = p1[1], z1 = p1[2];
  v2f a; a.x = hi ? z1 : x1; a.y = hi ? 0.0f : y1;
  const float n2 = x1 * x1 + y1 * y1 + z1 * z1;
  const float lf = Lfac[b * NN + row];
  float n2row[8], lfrow[8];
#pragma unroll
  for (int g = 0; g < 8; ++g) {
    n2row[g] = __shfl(n2, g + 8 * hi);
    lfrow[g] = __shfl(lf, g + 8 * hi);
  }

  float accW[8] = {0, 0, 0, 0, 0, 0, 0, 0};
  float accC[8] = {0, 0, 0, 0, 0, 0, 0, 0};
  const float* x2b = xyz2 + b * NN * 3;
  const float* rb  = remR + b * NN;
  const float* fb  = fac + b * NN;
  for (int t = 0; t < TILES; ++t) {
    const int col = t * 16 + lo;
    const float* p2 = x2b + col * 3;
    const float x2 = p2[0], y2 = p2[1], z2 = p2[2];
    const float m2 = x2 * x2 + y2 * y2 + z2 * z2;
    const float rr = rb[col];
    const float fc = fb[col];
    v2f bv; bv.x = hi ? z2 : x2; bv.y = hi ? 0.0f : y2;
    v8f d = dot_tile(a, bv);
#pragma unroll
    for (int g = 0; g < 8; ++g) {
      float d2 = fmaxf(n2row[g] + m2 - 2.0f * d[g], 0.0f);
      float w = __expf(level * d2) * rr * lfrow[g] * fc;
      accW[g] += w;
      accC[g] += w * sqrtf(d2);
    }
  }
#pragma unroll
  for (int g = 0; g < 8; ++g) {
    accW[g] += __shfl_xor(accW[g], 1);
    accW[g] += __shfl_xor(accW[g], 2);
    accW[g] += __shfl_xor(accW[g], 4);
    accW[g] += __shfl_xor(accW[g], 8);
    accC[g] += __shfl_xor(accC[g], 1);
    accC[g] += __shfl_xor(accC[g], 2);
    accC[g] += __shfl_xor(accC[g], 4);
    accC[g] += __shfl_xor(accC[g], 8);
  }
  if (lo == 0) {
#pragma unroll
    for (int g = 0; g < 8; ++g) {
      const int idx = b * NN + strip * 16 + g + 8 * hi;
      remL[idx] = fmaxf(remL[idx] - accW[g], 0.0f);
      costRow[idx] += accC[g];
    }
  }
}

// ---- Pass 3b: per-column w sums; remainR update (in place) ----------------
__global__ void __launch_bounds__(256)
k_update_cols(const float* __restrict__ xyz1, const float* __restrict__ xyz2,
              float* __restrict__ remR, const float* __restrict__ Lfac,
              const float* __restrict__ fac, float level) {
  const int wid  = (blockIdx.x * 256 + threadIdx.x) >> 5;
  const int lane = threadIdx.x & 31;
  const int lo = lane & 15, hi = lane >> 4;
  const int b = wid >> 7, strip = wid & 127;

  const int col = strip * 16 + lo;
  const float* p2 = xyz2 + (b * NN + col) * 3;
  const float x2 = p2[0], y2 = p2[1], z2 = p2[2];
  v2f bv; bv.x = hi ? z2 : x2; bv.y = hi ? 0.0f : y2;
  const float m2 = x2 * x2 + y2 * y2 + z2 * z2;
  const float rr = remR[b * NN + col];
  const float fc = fac[b * NN + col];

  float accW = 0.0f;
  const float* x1b = xyz1 + b * NN * 3;
  const float* lfb = Lfac + b * NN;
  for (int t = 0; t < TILES; ++t) {
    const int row = t * 16 + lo;
    const float* p1 = x1b + row * 3;
    const float x1 = p1[0], y1 = p1[1], z1 = p1[2];
    v2f a; a.x = hi ? z1 : x1; a.y = hi ? 0.0f : y1;
    const float n2 = x1 * x1 + y1 * y1 + z1 * z1;
    const float lf = lfb[row];
    float n2row[8], lfrow[8];
#pragma unroll
    for (int g = 0; g < 8; ++g) {
      n2row[g] = __shfl(n2, g + 8 * hi);
      lfrow[g] = __shfl(lf, g + 8 * hi);
    }
    v8f d = dot_tile(a, bv);
#pragma unroll
    for (int g = 0; g < 8; ++g) {
      float d2 = fmaxf(n2row[g] + m2 - 2.0f * d[g], 0.0f);
      accW += __expf(level * d2) * rr * lfrow[g] * fc;
    }
  }
  accW += __shfl_down(accW, 16);
  if (lane < 16) {
    const int idx = b * NN + strip * 16 + lane;
    remR[idx] = fmaxf(remR[idx] - accW, 0.0f);
  }
}

// ---- Elementwise helpers ---------------------------------------------------
__global__ void __launch_bounds__(256)
k_init(float* remL, float* remR, float* costRow) {
  const int i = blockIdx.x * 256 + threadIdx.x;
  if (i < BB * NN) { remL[i] = 1.0f; remR[i] = 1.0f; costRow[i] = 0.0f; }
}
__global__ void __launch_bounds__(256)
k_lfac(const float* remL, float* rs /*in: rowsum, out: Lfac*/) {
  const int i = blockIdx.x * 256 + threadIdx.x;
  if (i < BB * NN) rs[i] = remL[i] / (rs[i] + 1e-9f);
}
__global__ void __launch_bounds__(256)
k_factor(const float* remR, float* cs /*in: colsum, out: factor*/) {
  const int i = blockIdx.x * 256 + threadIdx.x;
  if (i < BB * NN) cs[i] = fminf(remR[i] / (cs[i] + 1e-9f), 1.0f);
}
__global__ void __launch_bounds__(256)
k_final(const float* __restrict__ costRow, float* __restrict__ out) {
  __shared__ float s[256];
  const int b = blockIdx.x;
  float v = 0.0f;
  for (int i = threadIdx.x; i < NN; i += 256) v += costRow[b * NN + i];
  s[threadIdx.x] = v;
  __syncthreads();
  for (int off = 128; off > 0; off >>= 1) {
    if (threadIdx.x < off) s[threadIdx.x] += s[threadIdx.x + off];
    __syncthreads();
  }
  if (threadIdx.x == 0) out[b] = s[0];
}

extern "C" void kernel_launch(void* const* d_in, const int* in_sizes, int n_in,
                              void* d_out, int out_size, void* d_ws, size_t ws_size,
                              hipStream_t stream) {
  const float* xyz1 = (const float*)d_in[0];
  const float* xyz2 = (const float*)d_in[1];
  float* ws = (float*)d_ws;
  const int A = BB * NN;
  float* remL    = ws + 0 * A;
  float* remR    = ws + 1 * A;
  float* rs      = ws + 2 * A;  // rowsum -> Lfac (in place)
  float* cs      = ws + 3 * A;  // colsum -> factor (in place)
  float* costRow = ws + 4 * A;  // 5 * 64KB = 320KB total workspace

  const dim3 blk(256);
  const dim3 gElem((A + 255) / 256);
  const dim3 gTile(STRIPS / 8);  // 8 waves/block, 1024 wave-strips

  k_init<<<gElem, blk, 0, stream>>>(remL, remR, costRow);

  const float levels[10] = {-16384.f, -4096.f, -1024.f, -256.f, -64.f,
                            -16.f,    -4.f,    -1.f,    -0.25f, 0.f};
  for (int it = 0; it < 10; ++it) {
    const float level = levels[it];
    k_rowsum<<<gTile, blk, 0, stream>>>(xyz1, xyz2, remR, rs, level);
    k_lfac<<<gElem, blk, 0, stream>>>(remL, rs);
    k_colsum<<<gTile, blk, 0, stream>>>(xyz1, xyz2, remR, rs, cs, level);
    k_factor<<<gElem, blk, 0, stream>>>(remR, cs);
    k_update_rows<<<gTile, blk, 0, stream>>>(xyz1, xyz2, remR, rs, cs, remL,
                                             costRow, level);
    k_update_cols<<<gTile, blk, 0, stream>>>(xyz1, xyz2, remR, rs, cs, level);
  }
  k_final<<<dim3(BB), blk, 0, stream>>>(costRow, (float*)d_out);
}